// Model_13855564497062
// MI455X (gfx1250) — compile-verified
//
#include <hip/hip_runtime.h>

typedef __attribute__((ext_vector_type(2))) float v2f;
typedef __attribute__((ext_vector_type(8))) float v8f;

// ---------------------------------------------------------------------------
// Input dense layer: h[0:32] = relu(x @ W_in + b_in), h[32:64] = 0 (ZeroPad).
// One wave32 computes a 16x32 output tile with K=64 using V_WMMA_F32_16X16X4_F32.
// A fragment (16x4 f32): lane l holds M=l%16, K = k0 + 2*(l/16) + {0,1}.
// B fragment (4x16 f32): lane l holds N=l%16, K = k0 + 2*(l/16) + {0,1}.
// C/D (16x16 f32, v8f):  lane l, reg r -> M = r + 8*(l/16), N = l%16.
// ---------------------------------------------------------------------------
__global__ __launch_bounds__(256) void gemm_relu_pad_kernel(
    const float* __restrict__ x, const float* __restrict__ W,
    const float* __restrict__ b, float* __restrict__ h, int N)
{
    int wave = (int)((blockIdx.x * blockDim.x + threadIdx.x) >> 5);
    int lane = threadIdx.x & 31;
    int row0 = wave * 16;
    if (row0 >= N) return;                 // wave-uniform exit (EXEC stays all-1)

    int m    = lane & 15;
    int half = lane >> 4;

    int arow = row0 + m;
    if (arow > N - 1) arow = N - 1;        // safety clamp (N%16==0 in practice)
    const float* __restrict__ ap = x + (size_t)arow * 64;

    v8f c0 = {};                           // N-tile 0 (cols 0..15)
    v8f c1 = {};                           // N-tile 1 (cols 16..31)

    #pragma unroll
    for (int k0 = 0; k0 < 64; k0 += 4) {
        int k = k0 + 2 * half;
        v2f a;
        a.x = ap[k];
        a.y = ap[k + 1];
        v2f b0, b1;
        b0.x = W[(size_t)k * 32 + m];
        b0.y = W[(size_t)(k + 1) * 32 + m];
        b1.x = W[(size_t)k * 32 + 16 + m];
        b1.y = W[(size_t)(k + 1) * 32 + 16 + m];
        c0 = __builtin_amdgcn_wmma_f32_16x16x4_f32(
                 false, a, false, b0, (short)0, c0, false, false);
        c1 = __builtin_amdgcn_wmma_f32_16x16x4_f32(
                 false, a, false, b1, (short)0, c1, false, false);
    }

    float bia0 = b[m];
    float bia1 = b[16 + m];
    #pragma unroll
    for (int r = 0; r < 8; ++r) {
        int row = row0 + r + 8 * half;
        if (row >= N) continue;
        float v0 = c0[r] + bia0;
        float v1 = c1[r] + bia1;
        float* hp = h + (size_t)row * 64;
        hp[m]      = v0 > 0.f ? v0 : 0.f;
        hp[16 + m] = v1 > 0.f ? v1 : 0.f;
        hp[32 + m] = 0.f;                  // ZeroPad2d columns 32..63
        hp[48 + m] = 0.f;
    }
}

__global__ void zero_kernel(float* __restrict__ p, long long n)
{
    long long i = (long long)blockIdx.x * blockDim.x + threadIdx.x;
    if (i < n) p[i] = 0.f;
}

__global__ void deg_kernel(const long long* __restrict__ dst,
                           float* __restrict__ deg, int E)
{
    int i = blockIdx.x * blockDim.x + threadIdx.x;
    if (i < E) unsafeAtomicAdd(&deg[dst[i]], 1.0f);
}

__global__ void inv_kernel(float* __restrict__ deg, int N)
{
    int i = blockIdx.x * blockDim.x + threadIdx.x;
    if (i < N) deg[i] = 1.0f / fmaxf(deg[i], 1.0f);
}

// One thread per (edge, 4-feature chunk): 16 threads cover an edge's 64-float
// row with a coalesced 256B read, then 4 native f32 global atomics each.
__global__ __launch_bounds__(256) void scatter_kernel(
    const float* __restrict__ h, const long long* __restrict__ src,
    const long long* __restrict__ dst, float* __restrict__ msg, int E)
{
    long long idx = (long long)blockIdx.x * blockDim.x + threadIdx.x;
    long long total = (long long)E * 16;
    if (idx >= total) return;
    int e = (int)(idx >> 4);
    int f = (int)(idx & 15) * 4;
    long long s = src[e];
    long long d = dst[e];
    const float4 v = *(const float4*)(h + (size_t)s * 64 + f);
    float* mp = msg + (size_t)d * 64 + f;
    unsafeAtomicAdd(mp + 0, v.x);
    unsafeAtomicAdd(mp + 1, v.y);
    unsafeAtomicAdd(mp + 2, v.z);
    unsafeAtomicAdd(mp + 3, v.w);
}

// h = (h + m * inv_deg) * 0.5 ; also re-zero m for the next step.
__global__ void update_kernel(float* __restrict__ h, float* __restrict__ msg,
                              const float* __restrict__ inv, long long total)
{
    long long i = (long long)blockIdx.x * blockDim.x + threadIdx.x;
    if (i < total) {
        float mm = msg[i];
        msg[i] = 0.f;
        h[i] = (h[i] + mm * inv[i >> 6]) * 0.5f;
    }
}

// Per-block partial column sums of h into g[64] via atomics.
__global__ __launch_bounds__(256) void readout_kernel(
    const float* __restrict__ h, float* __restrict__ g, int N)
{
    __shared__ float s[256];
    int f   = threadIdx.x & 63;
    int sub = threadIdx.x >> 6;            // 0..3
    float acc = 0.f;
    for (int node = blockIdx.x * 4 + sub; node < N; node += gridDim.x * 4)
        acc += h[(size_t)node * 64 + f];
    s[threadIdx.x] = acc;
    __syncthreads();
    if (threadIdx.x < 64) {
        float v = s[threadIdx.x] + s[threadIdx.x + 64] +
                  s[threadIdx.x + 128] + s[threadIdx.x + 192];
        unsafeAtomicAdd(&g[threadIdx.x], v);
    }
}

__global__ void final_kernel(const float* __restrict__ g,
                             const float* __restrict__ Wp,
                             const float* __restrict__ bp,
                             float* __restrict__ out, float invN)
{
    if (threadIdx.x == 0) {
        float acc = bp[0];
        #pragma unroll
        for (int n = 0; n < 64; ++n) acc += g[n] * invN * Wp[n];
        out[0] = acc;
    }
}

extern "C" void kernel_launch(void* const* d_in, const int* in_sizes, int n_in,
                              void* d_out, int out_size, void* d_ws, size_t ws_size,
                              hipStream_t stream)
{
    const float*     x      = (const float*)d_in[0];
    const long long* ei     = (const long long*)d_in[1];  // [2, E] row-major
    const float*     W_in   = (const float*)d_in[2];
    const float*     b_in   = (const float*)d_in[3];
    const float*     W_pred = (const float*)d_in[4];
    const float*     b_pred = (const float*)d_in[5];
    float*           out    = (float*)d_out;

    const int N = in_sizes[0] / 64;
    const int E = in_sizes[1] / 2;
    const long long HN = (long long)N * 64;

    float* h   = (float*)d_ws;     // [N,64]
    float* msg = h + HN;           // [N,64]
    float* deg = msg + HN;         // [N]    (becomes 1/max(deg,1))
    float* g   = deg + N;          // [64]

    // Zero m, deg, g (contiguous) every call: deterministic w.r.t. poisoning.
    const long long zn = HN + N + 64;
    zero_kernel<<<(int)((zn + 255) / 256), 256, 0, stream>>>(msg, zn);

    // Input layer with WMMA.
    {
        int waves  = (N + 15) / 16;
        int blocks = (waves + 7) / 8;      // 8 waves (256 threads) per block
        gemm_relu_pad_kernel<<<blocks, 256, 0, stream>>>(x, W_in, b_in, h, N);
    }

    // In-degree, then reciprocal of clamped degree.
    deg_kernel<<<(E + 255) / 256, 256, 0, stream>>>(ei + E, deg, E);
    inv_kernel<<<(N + 255) / 256, 256, 0, stream>>>(deg, N);

    // 4 message-passing rounds.
    const long long sc_total = (long long)E * 16;
    for (int step = 0; step < 4; ++step) {
        scatter_kernel<<<(int)((sc_total + 255) / 256), 256, 0, stream>>>(
            h, ei, ei + E, msg, E);
        update_kernel<<<(int)((HN + 255) / 256), 256, 0, stream>>>(
            h, msg, deg, HN);
    }

    // Readout: mean over nodes, then affine prediction.
    readout_kernel<<<256, 256, 0, stream>>>(h, g, N);
    final_kernel<<<1, 64, 0, stream>>>(g, W_pred, b_pred, out, 1.0f / (float)N);
}